// ROIPooling_16312285790450
// MI455X (gfx1250) — compile-verified
//
#include <hip/hip_runtime.h>
#include <math.h>

// Problem constants (from the reference setup)
#define OH_ 7
#define OW_ 7
#define H_ 64
#define W_ 64
#define C_ 128

// Tiling: channels processed in two halves of 64 (16 float4 chunks / pixel)
#define CH_HALF   64
#define CK_PER_PX 16                 // 64 channels / 4 per chunk
#define NX_MAX    64                 // max staged columns (full W)
#define NCHUNK    (NX_MAX * CK_PER_PX)  // 1024 chunks = 16 KB of float4
#define NTHREADS  256
#define K_ITERS   (NCHUNK / NTHREADS)   // 4 async b128 ops per thread per row

// ---- CDNA5 async global->LDS copy (16B) + asynccnt wait -------------------
// CDNA5 ISA: GLOBAL_LOAD_ASYNC_TO_LDS_B128, GV mode:
//   vdst = LDS byte address (VGPR), vaddr = 64-bit global address (VGPR pair)
__device__ __forceinline__ void async_copy16(const void* g, void* l) {
  unsigned lds = (unsigned)(unsigned long long)l;        // low 32b = LDS addr
  unsigned long long ga = (unsigned long long)g;
  asm volatile("global_load_async_to_lds_b128 %0, %1, off"
               :: "v"(lds), "v"(ga) : "memory");
}

// Inline-asm wait with a "memory" clobber: hard compiler barrier so the LDS
// reads that consume the staged row can never be scheduled above the wait.
#define WAIT_ASYNC(n) \
  asm volatile("s_wait_asynccnt %0" :: "n"(n) : "memory")

// ---- TF/JAX bin grid: ceil(start + (stop-start)*i/7), fp32, no FMA --------
// t = i/7 precomputed as IEEE-RN fp32 constants (identical to runtime division)
__device__ __forceinline__ void bin_bounds(int start, int stop, int i,
                                           int* s_out, int* e_out) {
#pragma clang fp contract(off)
  static const float T[8] = {0.0f,        1.0f / 7.0f, 2.0f / 7.0f,
                             3.0f / 7.0f, 4.0f / 7.0f, 5.0f / 7.0f,
                             6.0f / 7.0f, 1.0f};
  const float fs = (float)start;
  const float d  = (float)(stop - start);
  int s = (int)ceilf(fs + d * T[i]);
  int e = (int)ceilf(fs + d * T[i + 1]);
  if (s == e) s -= 1;   // duplicate-boundary fixup
  *s_out = s;
  *e_out = e;
}

__device__ __forceinline__ float4 max4(float4 a, float4 b) {
  return make_float4(fmaxf(a.x, b.x), fmaxf(a.y, b.y),
                     fmaxf(a.z, b.z), fmaxf(a.w, b.w));
}

// Issue exactly K_ITERS async b128 copies per thread for one feature-map row
// (column px clamped so the per-wave outstanding count is a constant 4).
__device__ __forceinline__ void stage_row(const float* __restrict__ row_base,
                                          int nx, int h, float4* buf, int tid) {
#pragma unroll
  for (int k = 0; k < K_ITERS; ++k) {
    const int j  = tid + k * NTHREADS;
    int px = j >> 4;                 // pixel within [x1c, x1c+nx)
    px = px < nx ? px : nx - 1;      // clamp: harmless duplicate load
    const int c4 = j & 15;           // float4 chunk within 64-channel half
    const float* g = row_base + (size_t)px * C_ + h * CH_HALF + c4 * 4;
    async_copy16((const void*)g, (void*)&buf[j]);
  }
}

__global__ __launch_bounds__(NTHREADS)
void roi_maxpool_kernel(const float* __restrict__ fmaps,
                        const float* __restrict__ rois,
                        float* __restrict__ out, int R) {
  __shared__ float4 stage[2][NCHUNK];   // 32 KB double buffer
  __shared__ float4 rowmax[NCHUNK];     // 16 KB

  const int tid = threadIdx.x;
  const int oh = blockIdx.x;   // 3D grid: no integer division in the decode
  const int r  = blockIdx.y;
  const int b  = blockIdx.z;

  const float4 roi = ((const float4*)rois)[b * R + r];
  const int x1 = (int)(roi.x * (float)W_);
  const int y1 = (int)(roi.y * (float)H_);
  const int x2 = (int)(roi.z * (float)W_);
  const int y2 = (int)(roi.w * (float)H_);

  int ys, ye;
  bin_bounds(y1, y2 + 1, oh, &ys, &ye);
  const int y_lo = max(ys, 0);
  const int y_hi = min(ye, H_);

  // Staged x window: union of all ow bins' valid columns = [x1, min(x2,W-1)]
  const int x1c = min(max(x1, 0), W_ - 1);
  const int xhi = max(min(x2, W_ - 1), x1c);
  const int nx  = xhi - x1c + 1;

  const float* fb = fmaps + (size_t)b * H_ * W_ * C_;
  const float NEG = -__builtin_inff();

  for (int h = 0; h < 2; ++h) {
    float4 acc[K_ITERS];
#pragma unroll
    for (int k = 0; k < K_ITERS; ++k)
      acc[k] = make_float4(NEG, NEG, NEG, NEG);

    if (y_lo < y_hi) {
      // prefetch first row
      stage_row(fb + ((size_t)y_lo * W_ + x1c) * C_, nx, h, stage[0], tid);
      int p = 0;
      for (int y = y_lo; y < y_hi; ++y) {
        if (y + 1 < y_hi) {
          // overlap: launch next row, then retire only the current row's 4 ops
          stage_row(fb + ((size_t)(y + 1) * W_ + x1c) * C_, nx, h,
                    stage[p ^ 1], tid);
          WAIT_ASYNC(K_ITERS);
        } else {
          WAIT_ASYNC(0);
        }
#pragma unroll
        for (int k = 0; k < K_ITERS; ++k) {
          const float4 v = stage[p][tid + k * NTHREADS];
          acc[k] = max4(acc[k], v);
        }
        p ^= 1;
      }
    }

    // Publish row-max (each thread owns chunks j = tid + k*256)
#pragma unroll
    for (int k = 0; k < K_ITERS; ++k)
      rowmax[tid + k * NTHREADS] = acc[k];
    __syncthreads();

    // Column max per (ow, chunk): 7 * 16 = 112 active lanes
    if (tid < OW_ * CK_PER_PX) {
      const int ow = tid / CK_PER_PX;
      const int c4 = tid % CK_PER_PX;
      int xs, xe;
      bin_bounds(x1, x2 + 1, ow, &xs, &xe);
      int xb = max(max(xs, 0), x1c);
      int xn = min(min(xe, W_), x1c + nx);
      float4 m = make_float4(NEG, NEG, NEG, NEG);
      for (int x = xb; x < xn; ++x)
        m = max4(m, rowmax[(x - x1c) * CK_PER_PX + c4]);
      float* op = out +
          ((((size_t)(b * R + r) * OH_ + oh) * OW_ + ow) * C_) +
          h * CH_HALF + c4 * 4;
      *(float4*)op = m;
    }
    __syncthreads();   // stage/rowmax reused by next channel half
  }
}

extern "C" void kernel_launch(void* const* d_in, const int* in_sizes, int n_in,
                              void* d_out, int out_size, void* d_ws, size_t ws_size,
                              hipStream_t stream) {
  const float* fmaps = (const float*)d_in[0];
  const float* rois  = (const float*)d_in[1];
  float* out = (float*)d_out;

  const int B = in_sizes[0] / (H_ * W_ * C_);
  const int R = in_sizes[1] / (B * 4);

  dim3 grid(OH_, R, B);
  roi_maxpool_kernel<<<grid, NTHREADS, 0, stream>>>(fmaps, rois, out, R);
}